// WeightedIntervalCrossEntropyLoss_14869176779174
// MI455X (gfx1250) — compile-verified
//
#include <hip/hip_runtime.h>

#define NBINS 64
#define DFEAT 16
#define BCE_EPS 1e-7f
#define MIN_WEIGHT 0.01f

typedef __attribute__((ext_vector_type(2))) float v2f;
typedef __attribute__((ext_vector_type(4))) float v4f;
typedef __attribute__((ext_vector_type(8))) float v8f;

// ---------------------------------------------------------------------------
// Wave-level sum using V_WMMA_F32_16X16X4_F32 (wave32, EXEC must be all 1s).
// A is 16x4 f32: lane m holds A[m][0] (v0) / A[m][1] (v1) for lanes 0-15,
// A[m][2]/A[m][3] for lanes 16-31. Put x in slot0, zero slot1; B = ones.
// D[m][n] = x_m + x_{m+16}. Summing the 8 C/D VGPRs per lane gives the
// half-wave sums (lanes 0-15: rows 0-7, lanes 16-31: rows 8-15); one
// half-swap add yields the full wave sum in every lane.
// ---------------------------------------------------------------------------
__device__ __forceinline__ float wave_sum32(float x) {
#if __has_builtin(__builtin_amdgcn_wmma_f32_16x16x4_f32)
    v2f a; a[0] = x;    a[1] = 0.0f;
    v2f b; b[0] = 1.0f; b[1] = 1.0f;
    v8f c = {};
    c = __builtin_amdgcn_wmma_f32_16x16x4_f32(
            /*neg_a=*/false, a, /*neg_b=*/false, b,
            /*c_mod=*/(short)0, c, /*reuse_a=*/false, /*reuse_b=*/false);
    float s = ((c[0] + c[1]) + (c[2] + c[3])) + ((c[4] + c[5]) + (c[6] + c[7]));
    s += __shfl_xor(s, 16, 32);
    return s;
#else
    #pragma unroll
    for (int off = 16; off >= 1; off >>= 1) x += __shfl_xor(x, off, 32);
    return x;
#endif
}

__global__ void wice_zero(float* ws) {
    if (threadIdx.x == 0) ws[0] = 0.0f;
}

__global__ __launch_bounds__(256) void wice_main(
    const float* __restrict__ y_pred,
    const float* __restrict__ y_true,
    const float* __restrict__ bin_weights,
    const float* __restrict__ cond,
    const float* __restrict__ bin_edges,
    float* __restrict__ ws_sum,
    int npoints)
{
    __shared__ float s_edges[NBINS];
    __shared__ float s_bw[NBINS];
    __shared__ float s_block;

    const int tid = threadIdx.x;
    if (tid < NBINS) {
        s_edges[tid] = bin_edges[tid];
        s_bw[tid]    = bin_weights[tid];
    }
    if (tid == 0) s_block = 0.0f;
    __syncthreads();

    const int i = blockIdx.x * 256 + tid;
    float acc = 0.0f;

    if (i < npoints) {
        // --- branchless searchsorted(bin_edges, c, side='left') over 64 edges
        const float c = cond[i];
        int cnt = 0;
        #pragma unroll
        for (int step = 64; step >= 1; step >>= 1) {
            int t = cnt + step;
            if (t <= NBINS && s_edges[t - 1] < c) cnt = t;
        }
        int idx = cnt < 1 ? 1 : (cnt > NBINS ? NBINS : cnt);   // clip to [1,64]
        const float w = fmaxf(s_bw[idx - 1], MIN_WEIGHT);

        // --- weighted BCE over 16 contiguous elements (log2 domain)
        const v4f* __restrict__ yp = (const v4f*)(y_pred + (size_t)i * DFEAT);
        const v4f* __restrict__ yt = (const v4f*)(y_true + (size_t)i * DFEAT);
        float s = 0.0f;
        #pragma unroll
        for (int q = 0; q < 4; ++q) {
            v4f p4 = __builtin_nontemporal_load(&yp[q]);   // read-once stream
            v4f t4 = __builtin_nontemporal_load(&yt[q]);
            #pragma unroll
            for (int e = 0; e < 4; ++e) {
                float p   = fminf(fmaxf(p4[e], BCE_EPS), 1.0f - BCE_EPS);
                float l2p = __builtin_amdgcn_logf(p);          // v_log_f32 = log2
                float l2q = __builtin_amdgcn_logf(1.0f - p);
                // y*log2(p) + (1-y)*log2(1-p) = l2q + y*(l2p - l2q)
                s += fmaf(t4[e], l2p - l2q, l2q);
            }
        }
        acc = w * s;   // sign + ln2 folded into the finalize scale
    }

    // --- hierarchical reduction: WMMA wave sum -> LDS atomic -> global atomic
    const float wsum = wave_sum32(acc);           // full wave executes (EXEC all 1s)
    if ((tid & 31) == 0) atomicAdd(&s_block, wsum);
    __syncthreads();
    if (tid == 0) atomicAdd(ws_sum, s_block);
}

__global__ void wice_finalize(const float* __restrict__ ws_sum,
                              float* __restrict__ out, float scale) {
    if (threadIdx.x == 0) out[0] = ws_sum[0] * scale;
}

extern "C" void kernel_launch(void* const* d_in, const int* in_sizes, int n_in,
                              void* d_out, int out_size, void* d_ws, size_t ws_size,
                              hipStream_t stream) {
    const float* y_pred      = (const float*)d_in[0];
    const float* y_true      = (const float*)d_in[1];
    const float* bin_weights = (const float*)d_in[2];
    const float* cond        = (const float*)d_in[3];
    const float* bin_edges   = (const float*)d_in[4];
    float* out = (float*)d_out;
    float* acc = (float*)d_ws;

    const int npoints = in_sizes[3];              // conditional_info length (B)
    // scale = -ln(2) / (B * D): converts log2 back to ln, applies BCE sign & mean
    const float scale = (float)(-0.69314718055994530942 /
                                ((double)npoints * (double)DFEAT));

    wice_zero<<<1, 32, 0, stream>>>(acc);
    const int blocks = (npoints + 255) / 256;
    wice_main<<<blocks, 256, 0, stream>>>(y_pred, y_true, bin_weights, cond,
                                          bin_edges, acc, npoints);
    wice_finalize<<<1, 32, 0, stream>>>(acc, out, scale);
}